// CartesianSHT_7799660609692
// MI455X (gfx1250) — compile-verified
//
#include <hip/hip_runtime.h>

typedef float v2f __attribute__((ext_vector_type(2)));
typedef float v8f __attribute__((ext_vector_type(8)));

#define NLAT  256
#define NLON  512
#define BE    4      // B*E
#define NV    64     // V
#define LMAX  256
#define MMAX  257
#define NCH   256    // BE*NV channels
#define RROWS 514    // 2*MMAX (interleaved re/im rows)
#define RPAD  544    // padded to 17 row-pair tiles of 32

// ---------------------------------------------------------------------------
// Kernel 1: build the scaled DFT matrix F[RPAD][NLON].
// Row r = 2*m + comp:  comp==0 ->  (2pi/512)*cos(2pi*m*j/512)
//                      comp==1 -> -(2pi/512)*sin(2pi*m*j/512)
// Rows >= RROWS are zero so stage-1 needs no bounds checks anywhere.
// ---------------------------------------------------------------------------
__global__ void sht_fill_dft(float* __restrict__ F) {
    int idx = blockIdx.x * blockDim.x + threadIdx.x;
    if (idx >= RPAD * NLON) return;
    int r = idx >> 9;          // /512
    int j = idx & (NLON - 1);
    float val = 0.0f;
    if (r < RROWS) {
        int m = r >> 1;
        int t = (m * j) & (NLON - 1);          // angle reduced mod 2*pi
        const float scale = 6.283185307179586f / 512.0f;
        float ang = (float)t * scale;
        float s, c;
        __sincosf(ang, &s, &c);
        val = (r & 1) ? -scale * s : scale * c;
    }
    F[idx] = val;
}

// ---------------------------------------------------------------------------
// Stage 1: DFT as WMMA f32 GEMM, 2 row-tiles per wave sharing each B fragment.
// For each (k, be):  C(544x64) = F(544x512) x X_k(512x64),
// where X_k[j][v] = data[be, k*NLON + j, v].
// Wave computes a 32x16 tile: per K-step, 2 A-pair loads (b64, L2-resident F),
// 2 B loads (b32) shared by both v_wmma_f32_16x16x4_f32.
// Xhat is padded to RPAD rows -> all stores unconditional.
// ---------------------------------------------------------------------------
__global__ void __launch_bounds__(256)
sht_stage1(const float* __restrict__ data,
           const float* __restrict__ F,
           float* __restrict__ Xhat) {
    const int lane = threadIdx.x & 31;
    const int wave = threadIdx.x >> 5;
    int tile = blockIdx.x * (blockDim.x >> 5) + wave;
    // tile = ((be*256 + k)*4 + vt)*17 + rp
    int rp = tile % 17;  tile /= 17;     // 17 row-pair tiles (RPAD/32)
    int vt = tile & 3;   tile >>= 2;     // 4 col tiles (64/16)
    int k  = tile & 255; tile >>= 8;     // latitude row
    int be = tile;                       // batch*ensemble
    const int r0 = rp * 32, v0 = vt * 16;
    const int lo = lane & 15, hi = lane >> 4;

    // A fragment (16x4 f32): lane (lo,hi), vgpr v -> A[row + lo][kk + 2*hi + v]
    const float* Arow0 = F + (size_t)(r0 + lo) * NLON + 2 * hi;
    const float* Arow1 = Arow0 + (size_t)16 * NLON;
    // B fragment (4x16 f32): lane (lo,hi), vgpr v -> X[kk + 2*hi + v][v0 + lo]
    const float* Bcol = data + ((size_t)(be * NLAT + k) * NLON + 2 * hi) * NV
                             + v0 + lo;

    v8f acc0 = {0.f, 0.f, 0.f, 0.f, 0.f, 0.f, 0.f, 0.f};
    v8f acc1 = {0.f, 0.f, 0.f, 0.f, 0.f, 0.f, 0.f, 0.f};
    for (int kk = 0; kk < NLON; kk += 4) {
        v2f b;
        b.x = Bcol[(size_t)kk * NV];
        b.y = Bcol[(size_t)(kk + 1) * NV];
        v2f a0 = *(const v2f*)(Arow0 + kk);        // contiguous pair -> b64
        v2f a1 = *(const v2f*)(Arow1 + kk);
        acc0 = __builtin_amdgcn_wmma_f32_16x16x4_f32(
            false, a0, false, b, (short)0, acc0, false, false);
        acc1 = __builtin_amdgcn_wmma_f32_16x16x4_f32(
            false, a1, false, b, (short)0, acc1, false, false);
    }

    // C/D layout: vgpr v, lane (lo,hi) -> row base + v + 8*hi, col v0 + lo
    const int col = be * NV + v0 + lo;
    const int rbase = r0 + 8 * hi;
#pragma unroll
    for (int v = 0; v < 8; ++v) {
        Xhat[((size_t)(rbase + v) * NLAT + k) * NCH + col]      = acc0[v];
        Xhat[((size_t)(rbase + v + 16) * NLAT + k) * NCH + col] = acc1[v];
    }
}

// ---------------------------------------------------------------------------
// Stage 2: Legendre contraction as WMMA f32 GEMM, 2 l-tiles per wave.
// For each r = 2m+comp:  C(256x256) = P_m(256x256) x Xhat_r(256x256),
// P_m[l][k] = pct_w[l, m, k].  Output scattered into interleaved complex
// (B,E,LMAX,MMAX,V) layout: float index 2*((((be*LMAX+l)*MMAX+m)*V)+v)+comp.
// ---------------------------------------------------------------------------
__global__ void __launch_bounds__(256)
sht_stage2(const float* __restrict__ pct,
           const float* __restrict__ Xhat,
           float* __restrict__ out) {
    const int lane = threadIdx.x & 31;
    const int wave = threadIdx.x >> 5;
    int tile = blockIdx.x * (blockDim.x >> 5) + wave;
    // tile = (r*16 + nt)*8 + lp
    int lp = tile & 7;  tile >>= 3;      // 8 l-pair tiles (256/32)
    int nt = tile & 15; tile >>= 4;      // 16 n tiles
    int r  = tile;                       // 0..513
    const int m = r >> 1, comp = r & 1;
    const int l0 = lp * 32, n0 = nt * 16;
    const int lo = lane & 15, hi = lane >> 4;

    const float* Arow0 = pct + ((size_t)(l0 + lo) * MMAX + m) * NLAT + 2 * hi;
    const float* Arow1 = Arow0 + (size_t)16 * MMAX * NLAT;
    const float* Bbase = Xhat + ((size_t)r * NLAT + 2 * hi) * NCH + n0 + lo;

    v8f acc0 = {0.f, 0.f, 0.f, 0.f, 0.f, 0.f, 0.f, 0.f};
    v8f acc1 = {0.f, 0.f, 0.f, 0.f, 0.f, 0.f, 0.f, 0.f};
    for (int kk = 0; kk < NLAT; kk += 4) {
        v2f b;
        b.x = Bbase[(size_t)kk * NCH];
        b.y = Bbase[(size_t)(kk + 1) * NCH];
        v2f a0 = *(const v2f*)(Arow0 + kk);        // contiguous pair -> b64
        v2f a1 = *(const v2f*)(Arow1 + kk);
        acc0 = __builtin_amdgcn_wmma_f32_16x16x4_f32(
            false, a0, false, b, (short)0, acc0, false, false);
        acc1 = __builtin_amdgcn_wmma_f32_16x16x4_f32(
            false, a1, false, b, (short)0, acc1, false, false);
    }

    const int n   = n0 + lo;
    const int be  = n >> 6;             // which (b,e)
    const int vch = n & 63;             // channel within V
    const int lbase = l0 + 8 * hi;
#pragma unroll
    for (int v = 0; v < 8; ++v) {
        size_t i0 = 2 * (((((size_t)be * LMAX + lbase + v) * MMAX + m) * NV)
                         + vch) + comp;
        size_t i1 = 2 * (((((size_t)be * LMAX + lbase + v + 16) * MMAX + m) * NV)
                         + vch) + comp;
        out[i0] = acc0[v];
        out[i1] = acc1[v];
    }
}

// ---------------------------------------------------------------------------
// Launch: fill DFT matrix -> stage1 DFT GEMM -> stage2 Legendre GEMM.
// Workspace: [ F : RPAD*NLON floats ][ Xhat : RPAD*NLAT*NCH floats ]
// = 1.1 MB + 142.6 MB.
// ---------------------------------------------------------------------------
extern "C" void kernel_launch(void* const* d_in, const int* in_sizes, int n_in,
                              void* d_out, int out_size, void* d_ws, size_t ws_size,
                              hipStream_t stream) {
    const float* data = (const float*)d_in[0];   // (B,E,NLAT*NLON,V) f32
    const float* pct  = (const float*)d_in[1];   // (LMAX,MMAX,NLAT) f32
    float* out  = (float*)d_out;                 // complex64 interleaved
    float* F    = (float*)d_ws;
    float* Xhat = F + (size_t)RPAD * NLON;

    const int nF = RPAD * NLON;
    sht_fill_dft<<<(nF + 255) / 256, 256, 0, stream>>>(F);

    // stage1 tiles: 17 * 4 * 256 * 4 = 69632 waves, 8 waves/block
    sht_stage1<<<69632 / 8, 256, 0, stream>>>(data, F, Xhat);

    // stage2 tiles: 514 * 16 * 8 = 65792 waves, 8 waves/block
    sht_stage2<<<65792 / 8, 256, 0, stream>>>(pct, Xhat, out);
}